// FastQuantumEvolution_53807350284456
// MI455X (gfx1250) — compile-verified
//
#include <hip/hip_runtime.h>
#include <cstdint>

// FastQuantumEvolution on MI455X (gfx1250, wave32).
//
// Sparse graph propagate (gather/scatter SpMM, D=32) — no dense tiles, so the
// CDNA5 feature that applies is the async Global->LDS path, not WMMA.
// D_FEAT (32) == wave32 width: lane == feature, so every per-edge gather and
// atomic-scatter is one fully coalesced 128B wave op. All scratch (~78 MB)
// fits in the 192 MB L2, so gathers + f32 atomics resolve at L2, not HBM.

#define D_FEAT 32
#define EPB    256   // edges staged per block (one index pair per thread)

// ---------------------------------------------------------------- deg init ---
__global__ void qe_deg_init(float* __restrict__ deg, float* __restrict__ gsum, int n) {
    int i = blockIdx.x * blockDim.x + threadIdx.x;
    if (i < n) deg[i] = 1.0f;           // self loop contributes 1 to every node
    if (i == 0) *gsum = 0.0f;           // zero the global |psi|^2 accumulator
}

// ------------------------------------------------------------- deg count ----
__global__ void qe_deg_count(const long long* __restrict__ col,
                             float* __restrict__ deg, int e) {
    int i = blockIdx.x * blockDim.x + threadIdx.x;
    if (i < e) atomicAdd(&deg[(int)col[i]], 1.0f);
}

// ------------------------------------------------------- deg^{-1/2} ----------
__global__ void qe_dis(const float* __restrict__ deg, float* __restrict__ dis, int n) {
    int i = blockIdx.x * blockDim.x + threadIdx.x;
    if (i < n) {
        float d = deg[i];
        dis[i] = (d > 0.0f) ? rsqrtf(d) : 0.0f;
    }
}

// ------------------------------- out = dis[n]^2 * h  (self-loop term + init) --
__global__ void qe_self_init(const float* __restrict__ hr, const float* __restrict__ hi,
                             const float* __restrict__ dis,
                             float* __restrict__ outr, float* __restrict__ outi,
                             int total) {  // total = N * D_FEAT
    int i = blockIdx.x * blockDim.x + threadIdx.x;
    if (i < total) {
        float s = dis[i >> 5];          // lane-uniform within a wave, L2 hit
        s *= s;
        outr[i] = hr[i] * s;
        outi[i] = hi[i] * s;
    }
}

// ---------------- edge scatter: out[row] += h[col] * dis[row]*dis[col] -------
// Each block stages a 256-edge tile of (row,col) int64 indices into LDS with
// gfx1250 async Global->LDS transfers (ASYNCcnt), then each of the 8 waves
// processes 32 edges with wave-uniform indices from LDS: per edge two
// coalesced 128B gathers + two coalesced 128B f32 atomic-adds (L2-resident).
__global__ __launch_bounds__(256) void qe_edge_prop(
        const long long* __restrict__ rows, const long long* __restrict__ cols,
        const float* __restrict__ dis,
        const float* __restrict__ hr, const float* __restrict__ hi,
        float* __restrict__ outr, float* __restrict__ outi, int e) {
    __shared__ long long s_row[EPB];
    __shared__ long long s_col[EPB];

    const int t    = threadIdx.x;
    const int base = blockIdx.x * EPB;

    // speculative prefetch of the NEXT block's index tile (global_prefetch_b8)
    __builtin_prefetch(rows + base + EPB + t, 0, 0);
    __builtin_prefetch(cols + base + EPB + t, 0, 0);

    // ---- async stage of this block's index tile into LDS ----
    int eclamp = base + t;
    if (eclamp > e - 1) eclamp = e - 1;                 // tail-safe (unused slots)
    unsigned goff = (unsigned)((size_t)eclamp * 8u);    // byte offset into int64 stream
    unsigned ldr  = (unsigned)(uintptr_t)&s_row[t];     // LDS byte address (low 32 bits)
    unsigned ldc  = (unsigned)(uintptr_t)&s_col[t];
    asm volatile("global_load_async_to_lds_b64 %0, %1, %2"
                 :: "v"(ldr), "v"(goff), "s"(rows) : "memory");
    asm volatile("global_load_async_to_lds_b64 %0, %1, %2"
                 :: "v"(ldc), "v"(goff), "s"(cols) : "memory");
    asm volatile("s_wait_asynccnt 0x0" ::: "memory");
    __syncthreads();

    const int wave  = t >> 5;
    const int lane  = t & 31;
    const int ebase = base + wave * 32;

    #pragma unroll 4
    for (int j = 0; j < 32; ++j) {
        int eg = ebase + j;
        if (eg >= e) break;
        int r = (int)s_row[wave * 32 + j];              // wave-uniform LDS read
        int c = (int)s_col[wave * 32 + j];
        float nrm = dis[r] * dis[c];
        int cb = c * D_FEAT + lane;
        int rb = r * D_FEAT + lane;
        atomicAdd(&outr[rb], hr[cb] * nrm);             // global_atomic_add_f32
        atomicAdd(&outi[rb], hi[cb] * nrm);
    }
}

// -------- combine Taylor terms, emit evolved (interleaved complex) + |psi|^2 --
// One wave per node (lane == feature); 32 waves per block; two-level reduction
// (wave shuffle -> LDS -> wave shuffle) so only 1 global atomic per 32 nodes.
__global__ __launch_bounds__(1024) void qe_finalize(
        const float* __restrict__ xr, const float* __restrict__ xi,
        const float* __restrict__ fr, const float* __restrict__ fi,
        const float* __restrict__ sr, const float* __restrict__ si,
        const float* __restrict__ tptr, const float* __restrict__ dptr,
        float* __restrict__ out, float* __restrict__ gsum, int n) {
    __shared__ float part[32];
    int wib  = threadIdx.x >> 5;        // wave in block
    int lane = threadIdx.x & 31;
    int node = blockIdx.x * 32 + wib;

    float a   = tptr[0] * dptr[0];      // coeff1 = i*a
    float a2h = 0.5f * a * a;           // (i*a)^2/2 = -a^2/2, applied with minus

    float mag = 0.0f;
    if (node < n) {
        int idx = node * D_FEAT + lane;
        // evolved = x + i*a*fo - (a^2/2)*so
        float er = xr[idx] - a * fi[idx] - a2h * sr[idx];
        float ei = xi[idx] + a * fr[idx] - a2h * si[idx];
        // evolved output: complex64 flat (re,im interleaved) after w region
        float2* ev = (float2*)(out + n);
        ev[idx] = make_float2(er, ei);
        mag = er * er + ei * ei;
    }
    // wave32 reduction over the 32 features
    #pragma unroll
    for (int off = 16; off > 0; off >>= 1) mag += __shfl_xor(mag, off, 32);
    if (lane == 0) {
        if (node < n) out[node] = mag;  // un-normalized w
        part[wib] = mag;                // node>=n waves contribute 0
    }
    __syncthreads();
    if (threadIdx.x < 32) {
        float v = part[threadIdx.x];
        #pragma unroll
        for (int off = 16; off > 0; off >>= 1) v += __shfl_xor(v, off, 32);
        if (threadIdx.x == 0) atomicAdd(gsum, v);
    }
}

// ------------------------------------------------ normalize w in place ------
__global__ void qe_norm(float* __restrict__ out, const float* __restrict__ gsum, int n) {
    int i = blockIdx.x * blockDim.x + threadIdx.x;
    if (i < n) {
        float s = *gsum;
        out[i] = (s > 1e-8f) ? out[i] * ((float)n / s) : 1.0f;
    }
}

extern "C" void kernel_launch(void* const* d_in, const int* in_sizes, int n_in,
                              void* d_out, int out_size, void* d_ws, size_t ws_size,
                              hipStream_t stream) {
    (void)n_in; (void)out_size; (void)ws_size;

    const float*     xr   = (const float*)d_in[0];          // [N, 32]
    const float*     xi   = (const float*)d_in[1];          // [N, 32]
    const long long* eidx = (const long long*)d_in[2];      // [2, E] int64
    const float*     tev  = (const float*)d_in[3];          // scalar
    const float*     dstr = (const float*)d_in[4];          // scalar

    const int N = in_sizes[0] / D_FEAT;                     // 100000
    const int E = in_sizes[2] / 2;                          // 1600000
    const long long* rows = eidx;
    const long long* cols = eidx + E;

    // Workspace layout (floats): deg[N] | dis[N] | fo_r[ND] | fo_i[ND] |
    //                            so_r[ND] | so_i[ND] | gsum[1]   (~52 MB)
    float* w    = (float*)d_ws;
    const size_t ND = (size_t)N * D_FEAT;
    float* deg  = w;
    float* dis  = deg + N;
    float* fo_r = dis + N;
    float* fo_i = fo_r + ND;
    float* so_r = fo_i + ND;
    float* so_i = so_r + ND;
    float* gsum = so_i + ND;

    float* out = (float*)d_out;   // [0,N): w   [N, N+2*ND): evolved (re,im)

    const int T = 256;
    const int nBlkN   = (N + T - 1) / T;
    const int nBlkE   = (E + T - 1) / T;
    const int nBlkND  = (int)((ND + T - 1) / T);
    const int nBlkTile = (E + EPB - 1) / EPB;               // 256-edge tiles
    const int nBlkFin  = (N + 31) / 32;                     // 32 node-waves / block

    // 1) degree with self loops; zero global sum
    qe_deg_init <<<nBlkN, T, 0, stream>>>(deg, gsum, N);
    qe_deg_count<<<nBlkE, T, 0, stream>>>(cols, deg, E);
    qe_dis      <<<nBlkN, T, 0, stream>>>(deg, dis, N);

    // 2) first_order = dis^2 * x  +  edge_scatter(x)
    qe_self_init<<<nBlkND, T, 0, stream>>>(xr, xi, dis, fo_r, fo_i, (int)ND);
    qe_edge_prop<<<nBlkTile, T, 0, stream>>>(rows, cols, dis, xr, xi, fo_r, fo_i, E);

    // 3) second_order = dis^2 * first_order + edge_scatter(first_order)
    qe_self_init<<<nBlkND, T, 0, stream>>>(fo_r, fo_i, dis, so_r, so_i, (int)ND);
    qe_edge_prop<<<nBlkTile, T, 0, stream>>>(rows, cols, dis, fo_r, fo_i, so_r, so_i, E);

    // 4) combine + |psi|^2 reduction, 5) normalize
    qe_finalize<<<nBlkFin, 1024, 0, stream>>>(xr, xi, fo_r, fo_i, so_r, so_i,
                                              tev, dstr, out, gsum, N);
    qe_norm    <<<nBlkN, T, 0, stream>>>(out, gsum, N);
}